// NashExpertRouter_74088185856333
// MI455X (gfx1250) — compile-verified
//
#include <hip/hip_runtime.h>
#include <hip/hip_bf16.h>

typedef __attribute__((ext_vector_type(2))) float v2f;
typedef __attribute__((ext_vector_type(8))) float v8f;

// ---- problem constants ----
static constexpr int D      = 384;          // embed dim
static constexpr int Hh     = 64;           // router hidden
static constexpr int NTOK   = 16 * 8192;    // 131072 tokens
static constexpr int WPAD   = 388;          // padded K-stride for W' in LDS (conflict-free)
static constexpr int WT_FLOATS    = 64 * WPAD;        // 24832
static constexpr int S1_OFF       = WT_FLOATS;        // 24832
static constexpr int B1_OFF       = S1_OFF + 64;      // 24896
static constexpr int STAGE_FLOATS = B1_OFF + 64;      // 24960 (copied ws -> LDS)
static constexpr int W2_OFF       = STAGE_FLOATS;     // LDS only: 256 floats
static constexpr int SMEM_FLOATS  = W2_OFF + 256;     // 25216
static constexpr int SMEM_BYTES   = SMEM_FLOATS * 4;  // 100864 B (< 320KB/WGP)

static constexpr int NBLK   = 512;
static constexpr int NWAVES = NBLK * 4;               // 2048 persistent waves
static constexpr int ACC_OFF = STAGE_FLOATS;          // ws: per-wave aux partials (8 each)

// output layout: probs | indices | aux | probs
static constexpr int IDX_OFF = NTOK * 4;              // 524288
static constexpr int AUX_OFF = IDX_OFF + NTOK;        // 655360
static constexpr int P2_OFF  = AUX_OFF + 1;           // 655361 (odd -> scalar stores)

// ---- prep: W'[n][k] = ln_w[k]*w1[k][n] (padded), s1[n], b1[n] ----
__global__ void router_prep(const float* __restrict__ ln_w, const float* __restrict__ ln_b,
                            const float* __restrict__ w1, float* __restrict__ ws) {
  int t = threadIdx.x;
  for (int idx = t; idx < WT_FLOATS; idx += 256) {
    int n = idx / WPAD;
    int k = idx - n * WPAD;
    ws[idx] = (k < D) ? ln_w[k] * w1[k * Hh + n] : 0.f;
  }
  if (t < Hh) {
    float a = 0.f, b = 0.f;
    for (int k = 0; k < D; ++k) {
      float w = w1[k * Hh + t];
      a += ln_w[k] * w;
      b += ln_b[k] * w;
    }
    ws[S1_OFF + t] = a;
    ws[B1_OFF + t] = b;
  }
}

__device__ __forceinline__ void ntile_step(const v8f& cv, const float* sS1, const float* sB1,
                                           const float* sW2, int n, const float rs[8],
                                           const float rb[8], float p[8][4]) {
  float s1n = sS1[n], b1n = sB1[n];
  const float4 wv = *(const float4*)(sW2 + n * 4);
#pragma unroll
  for (int j = 0; j < 8; ++j) {
    float hv = rs[j] * cv[j] - rb[j] * s1n + b1n;           // fused LayerNorm epilogue
    hv = 0.5f * hv * (1.f + erff(hv * 0.70710678118654752f)); // exact GELU
    p[j][0] += hv * wv.x;
    p[j][1] += hv * wv.y;
    p[j][2] += hv * wv.z;
    p[j][3] += hv * wv.w;
  }
}

__global__ __launch_bounds__(128, 1) void router_main(const float* __restrict__ x,
                                                      const float* __restrict__ w2g,
                                                      const float* __restrict__ ws,
                                                      float* __restrict__ out,
                                                      float* __restrict__ acc) {
  extern __shared__ float smem[];
  float* sW  = smem;                 // 64 x 388 fp32 W' (transposed, padded)
  float* sS1 = smem + S1_OFF;
  float* sB1 = smem + B1_OFF;
  float* sW2 = smem + W2_OFF;        // w2: [n][e] row-major, 256 floats

  const int tid = threadIdx.x;
  for (int i = tid; i < STAGE_FLOATS; i += 128) smem[i] = ws[i];
  for (int i = tid; i < 256; i += 128) sW2[i] = w2g[i];
  __syncthreads();

  const int wave = tid >> 5;
  const int lane = tid & 31;
  const int hi   = lane >> 4;        // K-half selector for A/B fragments
  const int lm   = lane & 15;        // row (A) / column (B,C,D) index
  const int gw   = blockIdx.x * 4 + wave;

  float aux8[8];
#pragma unroll
  for (int i = 0; i < 8; ++i) aux8[i] = 0.f;

  const float* bbase = sW + lm * WPAD + 2 * hi;

  for (int tile = gw; tile < NTOK / 16; tile += NWAVES) {
    const int t0 = tile * 16;
    const float* xrow = x + (size_t)(t0 + lm) * D + 2 * hi;

    v8f c0 = {}, c1 = {}, c2 = {}, c3 = {};
    float s = 0.f, ss = 0.f;

#pragma unroll 4
    for (int k0 = 0; k0 < D; k0 += 4) {
      v2f a = *(const v2f*)(xrow + k0);                 // A frag: x[row][k0+2hi .. +1]
      s  += a.x + a.y;                                  // LN stats from the same load
      ss += a.x * a.x + a.y * a.y;
      const float* bp = bbase + k0;
      v2f b0 = *(const v2f*)(bp);
      v2f b1 = *(const v2f*)(bp + 16 * WPAD);
      v2f b2 = *(const v2f*)(bp + 32 * WPAD);
      v2f b3 = *(const v2f*)(bp + 48 * WPAD);
      c0 = __builtin_amdgcn_wmma_f32_16x16x4_f32(false, a, false, b0, (short)0, c0, false, false);
      c1 = __builtin_amdgcn_wmma_f32_16x16x4_f32(false, a, false, b1, (short)0, c1, false, false);
      c2 = __builtin_amdgcn_wmma_f32_16x16x4_f32(false, a, false, b2, (short)0, c2, false, false);
      c3 = __builtin_amdgcn_wmma_f32_16x16x4_f32(false, a, false, b3, (short)0, c3, false, false);
    }

    // combine LN stats across the two K-halves of each row
    s  += __shfl_xor(s, 16);
    ss += __shfl_xor(ss, 16);
    float mu   = s * (1.f / 384.f);
    float var  = ss * (1.f / 384.f) - mu * mu;
    float rstd = rsqrtf(var + 1e-5f);
    float mr   = mu * rstd;

    float rs[8], rb[8];                                 // per-accumulator-row scale/shift
#pragma unroll
    for (int j = 0; j < 8; ++j) {
      rs[j] = __shfl(rstd, j + 8 * hi);
      rb[j] = __shfl(mr,   j + 8 * hi);
    }

    float p[8][4];
#pragma unroll
    for (int j = 0; j < 8; ++j)
#pragma unroll
      for (int e = 0; e < 4; ++e) p[j][e] = 0.f;

    ntile_step(c0, sS1, sB1, sW2, 0 * 16 + lm, rs, rb, p);
    ntile_step(c1, sS1, sB1, sW2, 1 * 16 + lm, rs, rb, p);
    ntile_step(c2, sS1, sB1, sW2, 2 * 16 + lm, rs, rb, p);
    ntile_step(c3, sS1, sB1, sW2, 3 * 16 + lm, rs, rb, p);

    // reduce logits partials over the 16 lanes of each half (sum over n=0..63)
#pragma unroll
    for (int j = 0; j < 8; ++j)
#pragma unroll
      for (int e = 0; e < 4; ++e) {
        float v = p[j][e];
        v += __shfl_xor(v, 1);
        v += __shfl_xor(v, 2);
        v += __shfl_xor(v, 4);
        v += __shfl_xor(v, 8);
        p[j][e] = v;
      }

    // lane lm (<8) of each half takes row j==lm  (token t0 + lm + 8*hi)
    float q0 = 0.f, q1 = 0.f, q2 = 0.f, q3 = 0.f;
#pragma unroll
    for (int j = 0; j < 8; ++j) {
      bool sel = (lm == j);
      q0 = sel ? p[j][0] : q0;
      q1 = sel ? p[j][1] : q1;
      q2 = sel ? p[j][2] : q2;
      q3 = sel ? p[j][3] : q3;
    }

    float l0 = q0 * 0.5f, l1 = q1 * 0.5f, l2 = q2 * 0.5f, l3 = q3 * 0.5f; // /T, T=2
    float m  = fmaxf(fmaxf(l0, l1), fmaxf(l2, l3));
    float e0 = expf(l0 - m), e1 = expf(l1 - m), e2 = expf(l2 - m), e3 = expf(l3 - m);
    float inv = 1.f / (e0 + e1 + e2 + e3);
    float pr0 = e0 * inv, pr1 = e1 * inv, pr2 = e2 * inv, pr3 = e3 * inv;

    int best = 0; float bv = l0;
    if (l1 > bv) { bv = l1; best = 1; }
    if (l2 > bv) { bv = l2; best = 2; }
    if (l3 > bv) { bv = l3; best = 3; }

    const bool act = (lm < 8);
    if (act) {
      size_t tg = (size_t)t0 + lm + 8 * hi;
      float4 pv; pv.x = pr0; pv.y = pr1; pv.z = pr2; pv.w = pr3;
      *(float4*)(out + tg * 4) = pv;
      out[P2_OFF + tg * 4 + 0] = pr0;
      out[P2_OFF + tg * 4 + 1] = pr1;
      out[P2_OFF + tg * 4 + 2] = pr2;
      out[P2_OFF + tg * 4 + 3] = pr3;
      out[IDX_OFF + tg] = (float)best;
    }

    // deterministic aux partials: fixed shuffle tree, per-wave register accum
    float add[8];
#pragma unroll
    for (int e = 0; e < 4; ++e) add[e] = (act && best == e) ? 1.f : 0.f;
    add[4] = act ? pr0 : 0.f;
    add[5] = act ? pr1 : 0.f;
    add[6] = act ? pr2 : 0.f;
    add[7] = act ? pr3 : 0.f;
#pragma unroll
    for (int i = 0; i < 8; ++i) {
      float v = add[i];
      v += __shfl_xor(v, 1);
      v += __shfl_xor(v, 2);
      v += __shfl_xor(v, 4);
      v += __shfl_xor(v, 8);
      v += __shfl_xor(v, 16);
      aux8[i] += v;
    }
  }

  if (lane == 0) {
#pragma unroll
    for (int i = 0; i < 8; ++i) acc[(size_t)gw * 8 + i] = aux8[i];
  }
}

__global__ void router_finalize(const float* __restrict__ acc, float* __restrict__ out) {
  float cnt[4] = {0.f, 0.f, 0.f, 0.f}, ps[4] = {0.f, 0.f, 0.f, 0.f};
  for (int w = 0; w < NWAVES; ++w) {
#pragma unroll
    for (int e = 0; e < 4; ++e) {
      cnt[e] += acc[w * 8 + e];
      ps[e]  += acc[w * 8 + 4 + e];
    }
  }
  const float inv = 1.f / (float)NTOK;
  float aux = 0.f;
#pragma unroll
  for (int e = 0; e < 4; ++e) aux += (cnt[e] * inv) * (ps[e] * inv);
  out[AUX_OFF] = 4.f * aux;
}

extern "C" void kernel_launch(void* const* d_in, const int* in_sizes, int n_in,
                              void* d_out, int out_size, void* d_ws, size_t ws_size,
                              hipStream_t stream) {
  const float* x   = (const float*)d_in[0];
  const float* lnw = (const float*)d_in[1];
  const float* lnb = (const float*)d_in[2];
  const float* w1  = (const float*)d_in[3];
  const float* w2  = (const float*)d_in[4];
  float* out = (float*)d_out;
  float* ws  = (float*)d_ws;

  router_prep<<<1, 256, 0, stream>>>(lnw, lnb, w1, ws);
  router_main<<<NBLK, 128, SMEM_BYTES, stream>>>(x, w2, ws, out, ws + ACC_OFF);
  router_finalize<<<1, 1, 0, stream>>>(ws + ACC_OFF, out);
}